// qualFace_52261162057740
// MI455X (gfx1250) — compile-verified
//
#include <hip/hip_runtime.h>
#include <math.h>

#define B_DIM 512
#define E_DIM 512
#define C_DIM 70722
#define OUT_SCM  ((size_t)B_DIM * C_DIM)      // scaled_cosine_m elements
#define M_CONST  0.4f
#define H_CONST  0.333f
#define S_CONST  64.0f
#define EPS_C    0.001f
#define PI_F     3.14159265358979323846f

typedef float v2f __attribute__((ext_vector_type(2)));
typedef float v8f __attribute__((ext_vector_type(8)));

// ---- monotonic float <-> uint encoding for atomicMax on floats ----
__device__ __forceinline__ unsigned int enc_f(float f) {
    unsigned int u = __float_as_uint(f);
    return (u & 0x80000000u) ? ~u : (u | 0x80000000u);
}
__device__ __forceinline__ float dec_f(unsigned int u) {
    return (u & 0x80000000u) ? __uint_as_float(u & 0x7FFFFFFFu)
                             : __uint_as_float(~u);
}
__device__ __forceinline__ float clampf(float x, float lo, float hi) {
    return fminf(fmaxf(x, lo), hi);
}

// ======================= 1) inverse column norms =======================
__global__ __launch_bounds__(256) void colnorm_kernel(
    const float* __restrict__ kmat, float* __restrict__ inv_cn) {
    int c = blockIdx.x * 256 + threadIdx.x;
    if (c >= C_DIM) return;
    float s = 0.0f;
    const float* p = kmat + c;
    #pragma unroll 4
    for (int e = 0; e < E_DIM; ++e) {
        float v = p[(size_t)e * C_DIM];
        s = fmaf(v, v, s);
    }
    inv_cn[c] = 1.0f / sqrtf(s);
}

// ======================= 2) per-row stats / heads =======================
__global__ __launch_bounds__(512) void stats_kernel(
    const float* __restrict__ emb, const float* __restrict__ norms,
    const int* __restrict__ label, const float* __restrict__ kmat,
    const float* __restrict__ inv_cn,
    const float* __restrict__ qual_w, const float* __restrict__ qual_b,
    const float* __restrict__ dist_w, const float* __restrict__ dist_b,
    float* __restrict__ g_ang, float* __restrict__ g_add,
    float* __restrict__ cos_label, float* __restrict__ dist_val,
    unsigned int* __restrict__ rowmax, float* __restrict__ q_loss_out) {
    __shared__ float ssn[B_DIM];
    __shared__ int   slab[B_DIM];
    __shared__ float red[B_DIM];
    const int b = threadIdx.x;

    float sn = clampf(norms[b], 0.001f, 100.0f);   // safe_norms
    int   lb = label[b];
    ssn[b] = sn; slab[b] = lb;
    __syncthreads();

    // sum reduction
    red[b] = sn; __syncthreads();
    for (int off = 256; off > 0; off >>= 1) {
        if (b < off) red[b] += red[b + off];
        __syncthreads();
    }
    float sum = red[0]; __syncthreads();
    // sum of squares
    red[b] = sn * sn; __syncthreads();
    for (int off = 256; off > 0; off >>= 1) {
        if (b < off) red[b] += red[b + off];
        __syncthreads();
    }
    float sumsq = red[0]; __syncthreads();

    float mean = sum * (1.0f / B_DIM);
    float var  = (sumsq - (float)B_DIM * mean * mean) / (float)(B_DIM - 1);
    float bs   = sqrtf(fmaxf(var, 0.0f));          // T_ALPHA == 1.0
    float bm   = mean;

    // per-class mean of safe_norms for this row's label (T_ALPHA==1 -> cwm = bgm)
    float csum = 0.0f, ccnt = 0.0f;
    for (int i = 0; i < B_DIM; ++i) {
        if (slab[i] == lb) { csum += ssn[i]; ccnt += 1.0f; }
    }
    float cwm = csum / fmaxf(ccnt, 1e-12f);

    float denom = bs + EPS_C;
    float ms    = clampf((sn - bm) / denom * H_CONST, -1.0f, 1.0f);
    float cwms  = clampf((sn - cwm) / denom, -1.0f, 1.0f) * H_CONST;
    g_ang[b] = -M_CONST * ms;
    g_add[b] =  M_CONST + M_CONST * ms;

    // qual / dist heads: dot(emb[b], w) + bias -> sigmoid
    const float* erow = emb + (size_t)b * E_DIM;
    float dq = 0.0f, dd = 0.0f;
    #pragma unroll 4
    for (int e = 0; e < E_DIM; ++e) {
        float x = erow[e];
        dq = fmaf(x, qual_w[e], dq);
        dd = fmaf(x, dist_w[e], dd);
    }
    float qual = 1.0f / (1.0f + expf(-(dq + qual_b[0])));
    float dst  = 1.0f / (1.0f + expf(-(dd + dist_b[0])));
    float qd = qual - (cwms + 1.0f) * 0.5f;
    q_loss_out[b] = qd * qd;
    dist_val[b]   = dst;

    // clipped cosine at the label column
    float dl = 0.0f;
    const float* kcol = kmat + lb;
    #pragma unroll 4
    for (int e = 0; e < E_DIM; ++e)
        dl = fmaf(erow[e], kcol[(size_t)e * C_DIM], dl);
    cos_label[b] = clampf(dl * inv_cn[lb], -1.0f + EPS_C, 1.0f - EPS_C);

    rowmax[b] = 0u;   // encoded floor (below enc(-2))
}

// =========== 3) WMMA f32 GEMM + margin epilogue + row-max ===========
// grid: x = ceil(C/128), y = 512/64. block = 256 threads (8 wave32).
__global__ __launch_bounds__(256) void gemm_margin_kernel(
    const float* __restrict__ emb, const float* __restrict__ kmat,
    const float* __restrict__ inv_cn, const int* __restrict__ label,
    const float* __restrict__ g_ang, const float* __restrict__ g_add,
    float* __restrict__ out, unsigned int* __restrict__ rowmax) {
    __shared__ unsigned int smax[64];
    __shared__ int          slab[64];

    const int tid  = threadIdx.x;
    const int lane = tid & 31;
    const int wave = tid >> 5;
    const int m_blk = blockIdx.y * 64;
    const long n_blk = (long)blockIdx.x * 128;

    if (tid < 64) {
        smax[tid] = 0u;
        slab[tid] = label[m_blk + tid];
    }
    __syncthreads();

    const int  wm = m_blk + (wave & 1) * 32;        // wave's 32 rows
    const long wn = n_blk + (long)(wave >> 1) * 32; // wave's 32 cols
    const int  r  = lane & 15;
    const int  kh = lane >> 4;

    // A fragment base pointers (16x4 f32 layout: lanes 0-15 K={0,1}, 16-31 K={2,3})
    const float* a0p = emb + (size_t)(wm + r) * E_DIM + 2 * kh;
    const float* a1p = emb + (size_t)(wm + 16 + r) * E_DIM + 2 * kh;
    // B fragment pointers (4x16 layout: row K striped across lanes in one VGPR)
    long n0 = wn + r;       if (n0 >= C_DIM) n0 = C_DIM - 1;
    long n1 = wn + 16 + r;  if (n1 >= C_DIM) n1 = C_DIM - 1;
    const float* b0p = kmat + (size_t)(2 * kh) * C_DIM + n0;
    const float* b1p = kmat + (size_t)(2 * kh) * C_DIM + n1;

    v8f acc00 = {}, acc01 = {}, acc10 = {}, acc11 = {};

    #pragma unroll 4
    for (int k0 = 0; k0 < E_DIM; k0 += 4) {
        v2f A0, A1, B0, B1;
        A0.x = a0p[0];      A0.y = a0p[1];
        A1.x = a1p[0];      A1.y = a1p[1];
        B0.x = b0p[0];      B0.y = b0p[C_DIM];
        B1.x = b1p[0];      B1.y = b1p[C_DIM];

        acc00 = __builtin_amdgcn_wmma_f32_16x16x4_f32(false, A0, false, B0,
                                                      (short)0, acc00, false, false);
        acc01 = __builtin_amdgcn_wmma_f32_16x16x4_f32(false, A0, false, B1,
                                                      (short)0, acc01, false, false);
        acc10 = __builtin_amdgcn_wmma_f32_16x16x4_f32(false, A1, false, B0,
                                                      (short)0, acc10, false, false);
        acc11 = __builtin_amdgcn_wmma_f32_16x16x4_f32(false, A1, false, B1,
                                                      (short)0, acc11, false, false);

        a0p += 4; a1p += 4;
        b0p += 4 * (size_t)C_DIM; b1p += 4 * (size_t)C_DIM;
    }

    v8f accarr[2][2] = { { acc00, acc01 }, { acc10, acc11 } };

    // Epilogue: normalize + clip + margin(label col) + scale, fused row-max
    for (int mi = 0; mi < 2; ++mi) {
        const int rowbase = wm + mi * 16 + kh * 8;  // first of this half-lane's 8 rows
        float vmax[8];
        #pragma unroll
        for (int j = 0; j < 8; ++j) vmax[j] = -2.0f;

        for (int ni = 0; ni < 2; ++ni) {
            long n = wn + ni * 16 + r;
            bool valid = (n < C_DIM);
            float icn = valid ? inv_cn[n] : 0.0f;
            if (valid) {
                #pragma unroll
                for (int j = 0; j < 8; ++j) {
                    int row = rowbase + j;
                    float c = clampf(accarr[mi][ni][j] * icn,
                                     -1.0f + EPS_C, 1.0f - EPS_C);
                    float o;
                    if ((int)n == slab[row - m_blk]) {
                        float th = clampf(acosf(c) + g_ang[row],
                                          EPS_C, PI_F - EPS_C);
                        o = (cosf(th) - g_add[row]) * S_CONST;
                    } else {
                        o = c * S_CONST;                 // cos(acos(c)) == c here
                        vmax[j] = fmaxf(vmax[j], c);
                    }
                    out[(size_t)row * C_DIM + n] = o;
                }
            }
        }
        // butterfly max across the 16 lanes of each half-wave (same row, diff n)
        #pragma unroll
        for (int j = 0; j < 8; ++j) {
            float v = vmax[j];
            for (int mask = 1; mask <= 8; mask <<= 1)
                v = fmaxf(v, __shfl_xor(v, mask, 32));
            if (r == 0)
                atomicMax(&smax[rowbase + j - m_blk], enc_f(v));
        }
    }
    __syncthreads();
    if (tid < 64) atomicMax(&rowmax[m_blk + tid], smax[tid]);
}

// ======================= 4) dist_est_loss =======================
__global__ __launch_bounds__(512) void finish_kernel(
    const float* __restrict__ cos_label, const float* __restrict__ dist_val,
    const unsigned int* __restrict__ rowmax, float* __restrict__ dist_loss_out) {
    int b = threadIdx.x;
    float rm  = dec_f(rowmax[b]);
    float tgt = cos_label[b] / (rm + EPS_C);
    float d   = dist_val[b] - tgt;
    dist_loss_out[b] = d * d;
}

extern "C" void kernel_launch(void* const* d_in, const int* in_sizes, int n_in,
                              void* d_out, int out_size, void* d_ws, size_t ws_size,
                              hipStream_t stream) {
    const float* emb    = (const float*)d_in[0];
    const float* norms  = (const float*)d_in[1];
    const int*   label  = (const int*)  d_in[2];
    const float* kmat   = (const float*)d_in[3];
    const float* qual_w = (const float*)d_in[4];
    const float* qual_b = (const float*)d_in[5];
    const float* dist_w = (const float*)d_in[6];
    const float* dist_b = (const float*)d_in[7];
    // d_in[8..10] (batch_mean/std, classwise_mean) unused: T_ALPHA == 1.0

    float* out = (float*)d_out;
    float* q_loss_out    = out + OUT_SCM;
    float* dist_loss_out = out + OUT_SCM + B_DIM;

    // workspace carve-up
    float* ws = (float*)d_ws;
    float* inv_cn    = ws;                 // C_DIM
    float* g_ang     = inv_cn + C_DIM;     // B
    float* g_add     = g_ang + B_DIM;      // B
    float* cos_label = g_add + B_DIM;      // B
    float* dist_val  = cos_label + B_DIM;  // B
    unsigned int* rowmax = (unsigned int*)(dist_val + B_DIM); // B

    colnorm_kernel<<<(C_DIM + 255) / 256, 256, 0, stream>>>(kmat, inv_cn);

    stats_kernel<<<1, B_DIM, 0, stream>>>(emb, norms, label, kmat, inv_cn,
                                          qual_w, qual_b, dist_w, dist_b,
                                          g_ang, g_add, cos_label, dist_val,
                                          rowmax, q_loss_out);

    dim3 grid((C_DIM + 127) / 128, B_DIM / 64);
    gemm_margin_kernel<<<grid, 256, 0, stream>>>(emb, kmat, inv_cn, label,
                                                 g_ang, g_add, out, rowmax);

    finish_kernel<<<1, B_DIM, 0, stream>>>(cos_label, dist_val, rowmax,
                                           dist_loss_out);
}